// QuantumMLP_27324581937502
// MI455X (gfx1250) — compile-verified
//
#include <hip/hip_runtime.h>

// ---------------------------------------------------------------------------
// 16-qubit batched state-vector simulator for MI455X (gfx1250, wave32).
// Gates of a layer are grouped 4 qubits at a time into 16x16 complex matrices
// and applied as 16x16x16 GEMMs via V_WMMA_F32_16X16X4_F32 (4 K-chunks).
// Two axes are fused per pass (per-wave LDS transpose between them). CNOT
// rings are GF(2)-linear index permutations fused into the loads of the next
// pass. State: SoA fp32 re/im planes, ping-pong in d_ws (HBM-bound ~12.8 GB).
// Tile gathers are software-pipelined with GLOBAL_LOAD_ASYNC_TO_LDS_B32
// (ASYNCcnt) when the toolchain exposes the builtin; direct loads otherwise.
// ---------------------------------------------------------------------------

typedef float v2f __attribute__((ext_vector_type(2)));
typedef float v8f __attribute__((ext_vector_type(8)));

#define NQ    16
#define DIM   65536u
#define BATCH 1024

#if defined(__has_builtin)
#  if __has_builtin(__builtin_amdgcn_global_load_async_to_lds_b32)
#    define QSIM_ASYNC 1
#  else
#    define QSIM_ASYNC 0
#  endif
#else
#  define QSIM_ASYNC 0
#endif

#if QSIM_ASYNC
#if __has_builtin(__builtin_amdgcn_s_wait_asynccnt)
#define QSIM_WAIT_ASYNC(n) __builtin_amdgcn_s_wait_asynccnt(n)
#else
#define QSIM_WAIT_ASYNC(n) asm volatile("s_wait_asynccnt %0" :: "i"(n) : "memory")
#endif
typedef __attribute__((address_space(1))) int g_i32;
typedef __attribute__((address_space(3))) int l_i32;
// Per-lane gather: copies 4 bytes from the global address g to the LDS address l; tracked by ASYNCcnt, completes in order.
__device__ inline void qsim_async_g32(const float* g, float* l) {
    __builtin_amdgcn_global_load_async_to_lds_b32((g_i32*)(g), (l_i32*)(l), 0, 0);
}
#endif

struct c32 { float re, im; };
__device__ inline c32 cmul(c32 a, c32 b) {
    return { a.re * b.re - a.im * b.im, a.re * b.im + a.im * b.re };
}

// Composite CNOT-ring permutation (ring shift R). Reference applies, for
// q = 0..15 in order: state = state[perm_q]. Load-gather composite
// fwd = P0∘P1∘...∘P15 (P15 innermost). Each P is a GF(2)-linear involution,
// so the composite is linear -> fusable into addresses by XOR decomposition.
template<int R>
__device__ inline unsigned perm_fwd(unsigned x) {
    if constexpr (R != 0) {
#pragma unroll
        for (int q = 15; q >= 0; --q) {
            unsigned cpos = 15u - (unsigned)q;
            unsigned tpos = 15u - (unsigned)((q + R) & 15);
            x ^= ((x >> cpos) & 1u) << tpos;
        }
    }
    return x;
}
template<int R>
__device__ inline unsigned perm_inv(unsigned x) {
    if constexpr (R != 0) {
#pragma unroll
        for (int q = 0; q <= 15; ++q) {
            unsigned cpos = 15u - (unsigned)q;
            unsigned tpos = 15u - (unsigned)((q + R) & 15);
            x ^= ((x >> cpos) & 1u) << tpos;
        }
    }
    return x;
}

// Build the four 2x2 gates for the 4-qubit group whose axis digit sits at
// index-bit shift S. Digit bit j corresponds to qubit q = 15 - S - j.
template<bool REAL>
__device__ inline void make_gates(int S, unsigned b,
                                  const float* __restrict__ x,
                                  const float* __restrict__ w,
                                  int rep, int layer, c32 (&u)[4][2][2]) {
#pragma unroll
    for (int j = 0; j < 4; ++j) {
        int q = 15 - S - j;
        if constexpr (REAL) {                  // RY(theta), theta per batch elem
            float th = 0.5f * x[b * NQ + q];
            float c = cosf(th), s = sinf(th);
            u[j][0][0] = { c, 0.f };  u[j][0][1] = { -s, 0.f };
            u[j][1][0] = { s, 0.f };  u[j][1][1] = {  c, 0.f };
        } else {                               // Rot(phi,theta,omega), shared
            const float* ww = w + (((rep * 2 + layer) * NQ) + q) * 3;
            float phi = ww[0], th = 0.5f * ww[1], om = ww[2];
            float c = cosf(th), s = sinf(th);
            float a = 0.5f * (phi + om), d = 0.5f * (phi - om);
            float ca = cosf(a), sa = sinf(a), cd = cosf(d), sd = sinf(d);
            u[j][0][0] = {  c * ca, -c * sa };   // exp(-i a) * c
            u[j][0][1] = { -s * cd, -s * sd };   // -exp(+i d) * s
            u[j][1][0] = {  s * cd, -s * sd };   //  exp(-i d) * s
            u[j][1][1] = {  c * ca,  c * sa };   //  exp(+i a) * c
        }
    }
}

// Kron-product entry U[M][K] = prod_j u[j][M_j][K_j]
__device__ inline c32 kron_entry(const c32 (&u)[4][2][2], unsigned M, unsigned K) {
    c32 p = { 1.f, 0.f };
#pragma unroll
    for (int j = 0; j < 4; ++j)
        p = cmul(p, u[j][(M >> j) & 1u][(K >> j) & 1u]);
    return p;
}

#define QWMMA(A, B, C) \
    __builtin_amdgcn_wmma_f32_16x16x4_f32(false, (A), false, (B), (short)0, (C), false, false)

// One streaming pass: apply the 16x16 group matrix on axis-shift SX, then on
// axis-shift SY, over 16x16 tiles. One workgroup (8 waves) per batch element,
// one tile per wave per iteration (32 iterations). Permutation R fused into
// load addresses. No cross-wave barriers: all LDS slices are wave-private and
// the compiler inserts the same-wave DScnt waits.
template<int SX, int SY, bool REAL, int R, bool INIT>
__global__ __launch_bounds__(256) void qsim_pass(
        const float* __restrict__ in_re, const float* __restrict__ in_im,
        float* __restrict__ out_re, float* __restrict__ out_im,
        const float* __restrict__ x, const float* __restrict__ w,
        int rep, int layer) {
    constexpr int BASE_SHIFT = (SX == 0) ? 8 : 0;  // where the 8 "other" bits live
    const unsigned b    = blockIdx.x;
    const unsigned tid  = threadIdx.x;
    const unsigned lane = tid & 31u;
    const unsigned wid  = tid >> 5;
    const unsigned h    = lane >> 4;      // K-pair half: K bit 1 (ISA A/B layout)
    const unsigned mn   = lane & 15u;     // M for A-operand, N for B-operand
    const size_t   bofs = (size_t)b << 16;

    // --- per-pass A-operands for both stages (constant across tiles) ---
    c32 U1[4][2][2], U2[4][2][2];
    make_gates<REAL>(SX, b, x, w, rep, layer, U1);
    make_gates<REAL>(SY, b, x, w, rep, layer, U2);
    v2f a1r[4], a1i[4], a1n[4], a2r[4], a2i[4], a2n[4];
#pragma unroll
    for (int kc = 0; kc < 4; ++kc) {
#pragma unroll
        for (int d = 0; d < 2; ++d) {
            unsigned K = 4u * kc + d + 2u * h;       // K = 4kc + comp + 2*half
            c32 e1 = kron_entry(U1, mn, K);
            c32 e2 = kron_entry(U2, mn, K);
            a1r[kc][d] = e1.re;  a1i[kc][d] = e1.im;
            a2r[kc][d] = e2.re;  a2i[kc][d] = e2.im;
        }
        a1n[kc] = -a1i[kc];   // f32 WMMA has no A-neg modifier: negate in VALU
        a2n[kc] = -a2i[kc];
    }

    // --- fused-permutation address pieces (linearity of the GF(2) map) ---
    unsigned pfk[4][2];
#pragma unroll
    for (int kc = 0; kc < 4; ++kc)
#pragma unroll
        for (int d = 0; d < 2; ++d)
            pfk[kc][d] = perm_fwd<R>((4u * kc + d + 2u * h) << SX);
    const unsigned pfn = perm_fwd<R>(mn << SY);

    __shared__ float lds[8][2][16 * 17];   // per-wave transpose slice (padded)
#if QSIM_ASYNC
    __shared__ float stage[8][2][2][256];  // per-wave double-buffered gather stage
#endif
    float* Lre = &lds[wid][0][0];
    float* Lim = &lds[wid][1][0];
    const v8f z = { 0.f, 0.f, 0.f, 0.f, 0.f, 0.f, 0.f, 0.f };

#if QSIM_ASYNC
    // Issue tile j's 16 per-lane gathers into stage buffer j&1. Slot layout
    // (kc*64 + 2*lane + d) makes the B-operand read back a single ds_load_b64.
    auto issue_tile = [&](unsigned j) {
        const unsigned t2   = wid + 8u * j;
        const unsigned pfb2 = perm_fwd<R>(t2 << BASE_SHIFT);
        float* sre = &stage[wid][j & 1u][0][0];
        float* sim = &stage[wid][j & 1u][1][0];
#pragma unroll
        for (int kc = 0; kc < 4; ++kc)
#pragma unroll
            for (int d = 0; d < 2; ++d) {
                unsigned idx = pfb2 ^ pfn ^ pfk[kc][d];
                qsim_async_g32(in_re + bofs + idx, sre + kc * 64 + 2 * lane + d);
                qsim_async_g32(in_im + bofs + idx, sim + kc * 64 + 2 * lane + d);
            }
    };
    if constexpr (!INIT) {        // pipeline prologue: tiles 0 and 1 in flight
        issue_tile(0);
        issue_tile(1);
    }
#endif

    for (unsigned it = 0; it < 32u; ++it) {
        const unsigned t    = wid + 8u * it;
        const unsigned base = t << BASE_SHIFT;

        // ---- B1[K=axisX digit][N=axisY digit] ----
        v2f bR[4], bI[4];
        if constexpr (INIT) {                 // virtual |0...0> initial state
#pragma unroll
            for (int kc = 0; kc < 4; ++kc)
#pragma unroll
                for (int d = 0; d < 2; ++d) {
                    unsigned idx = base ^ pfn ^ pfk[kc][d];
                    bR[kc][d] = (idx == 0u) ? 1.f : 0.f;
                    bI[kc][d] = 0.f;
                }
        } else {
#if QSIM_ASYNC
            if (it < 31u) { QSIM_WAIT_ASYNC(16); }  // tile `it` landed, it+1 in flight
            else          { QSIM_WAIT_ASYNC(0);  }
            asm volatile("" ::: "memory");          // fence: no LDS-read hoist
            const float* sre = &stage[wid][it & 1u][0][0];
            const float* sim = &stage[wid][it & 1u][1][0];
#pragma unroll
            for (int kc = 0; kc < 4; ++kc) {
                bR[kc] = *(const v2f*)(sre + kc * 64 + 2 * lane);
                bI[kc] = *(const v2f*)(sim + kc * 64 + 2 * lane);
            }
#else
            const unsigned pfb = perm_fwd<R>(base);
#pragma unroll
            for (int kc = 0; kc < 4; ++kc)
#pragma unroll
                for (int d = 0; d < 2; ++d) {
                    unsigned idx = pfb ^ pfn ^ pfk[kc][d];
                    bR[kc][d] = in_re[bofs + idx];
                    bI[kc][d] = in_im[bofs + idx];
                }
#endif
        }

        // ---- stage 1: D1 = U_X * B1 (complex via 2-4 real WMMAs / chunk) ----
        v8f dre = z, dim_ = z;
#pragma unroll
        for (int kc = 0; kc < 4; ++kc) {
            dre  = QWMMA(a1r[kc], bR[kc], dre);
            dim_ = QWMMA(a1r[kc], bI[kc], dim_);
            if constexpr (!REAL) {
                dre  = QWMMA(a1n[kc], bI[kc], dre);   // -Im(A)*Im(B)
                dim_ = QWMMA(a1i[kc], bR[kc], dim_);  //  Im(A)*Re(B)
            }
        }

#if QSIM_ASYNC
        if constexpr (!INIT) {     // refill the buffer we just drained
            asm volatile("" ::: "memory");
            if (it + 2u < 32u) issue_tile(it + 2u);
        }
#endif

        // ---- transpose D1 through wave-private LDS:
        //      B2[K=axisY][N=newX] = D1[newX][axisY] ----
#pragma unroll
        for (int j = 0; j < 8; ++j) {
            unsigned m = (unsigned)j + 8u * h;        // C/D layout: M = j (+8 hi half)
            Lre[m * 17u + mn] = dre[j];
            Lim[m * 17u + mn] = dim_[j];
        }
        v2f b2R[4], b2I[4];
#pragma unroll
        for (int kc = 0; kc < 4; ++kc) {
#pragma unroll
            for (int d = 0; d < 2; ++d) {
                unsigned K = 4u * kc + d + 2u * h;
                b2R[kc][d] = Lre[mn * 17u + K];
                b2I[kc][d] = Lim[mn * 17u + K];
            }
        }

        // ---- stage 2: D2 = U_Y * B2 ----
        v8f d2re = z, d2im = z;
#pragma unroll
        for (int kc = 0; kc < 4; ++kc) {
            d2re = QWMMA(a2r[kc], b2R[kc], d2re);
            d2im = QWMMA(a2r[kc], b2I[kc], d2im);
            if constexpr (!REAL) {
                d2re = QWMMA(a2n[kc], b2I[kc], d2re);
                d2im = QWMMA(a2i[kc], b2R[kc], d2im);
            }
        }

        // ---- store: index = base ^ (M<<SY) ^ (N<<SX), no perm on writes ----
#pragma unroll
        for (int j = 0; j < 8; ++j) {
            unsigned m   = (unsigned)j + 8u * h;
            unsigned idx = base ^ (m << SY) ^ (mn << SX);
            out_re[bofs + idx] = d2re[j];
            out_im[bofs + idx] = d2im[j];
        }
    }
}

// Z-expectations on all 16 wires; final CNOT-ring perm fused via inverse map.
// Deterministic tree reduction (no float atomics).
template<int R>
__global__ __launch_bounds__(256) void qsim_meas(
        const float* __restrict__ re, const float* __restrict__ im,
        float* __restrict__ out) {
    const unsigned b   = blockIdx.x;
    const unsigned tid = threadIdx.x;
    const size_t bofs  = (size_t)b << 16;

    float acc[16];
#pragma unroll
    for (int wq = 0; wq < 16; ++wq) acc[wq] = 0.f;

    for (unsigned s = tid; s < DIM; s += 256u) {
        float r = re[bofs + s], i2 = im[bofs + s];
        float p = r * r + i2 * i2;
        unsigned ilog = perm_inv<R>(s);   // logical index after the final perm
#pragma unroll
        for (int wq = 0; wq < 16; ++wq) {
            unsigned bit = (ilog >> (15 - wq)) & 1u;
            acc[wq] += __uint_as_float(__float_as_uint(p) ^ (bit << 31));
        }
    }

    __shared__ float red[256][17];
#pragma unroll
    for (int wq = 0; wq < 16; ++wq) red[tid][wq] = acc[wq];
    __syncthreads();
    for (unsigned st = 128u; st > 0u; st >>= 1) {
        if (tid < st)
#pragma unroll
            for (int wq = 0; wq < 16; ++wq) red[tid][wq] += red[tid + st][wq];
        __syncthreads();
    }
    if (tid < 16u) out[b * 16u + tid] = red[0][tid];
}

extern "C" void kernel_launch(void* const* d_in, const int* in_sizes, int n_in,
                              void* d_out, int out_size, void* d_ws, size_t ws_size,
                              hipStream_t stream) {
    const float* x = (const float*)d_in[0];      // (1024, 16) f32
    const float* w = (const float*)d_in[1];      // (2, 2, 16, 3) f32
    float* out = (float*)d_out;                  // (1024, 16) f32

    const size_t plane = (size_t)BATCH * DIM;    // floats per re/im plane
    if (ws_size < 4 * plane * sizeof(float)) return;  // need 1 GiB ping-pong
    float* Are = (float*)d_ws;
    float* Aim = Are + plane;
    float* Bre = Aim + plane;
    float* Bim = Bre + plane;

    dim3 g(BATCH), blk(256);
#define LP(SX, SY, RL, RR, IN, ir, ii, orr, oi, rp, ly) \
    qsim_pass<SX, SY, RL, RR, IN><<<g, blk, 0, stream>>>(ir, ii, orr, oi, x, w, rp, ly)

    // rep 0
    LP(0, 4,  true,  0, true,  (const float*)nullptr, (const float*)nullptr, Are, Aim, 0, 0); // RY lo-groups, virtual |0>
    LP(8, 12, true,  0, false, Are, Aim, Bre, Bim, 0, 0);  // RY hi-groups
    LP(0, 4,  false, 0, false, Bre, Bim, Are, Aim, 0, 0);  // Rot L0
    LP(8, 12, false, 0, false, Are, Aim, Bre, Bim, 0, 0);
    LP(0, 4,  false, 1, false, Bre, Bim, Are, Aim, 0, 1);  // Rot L1 (+perm r=1 on loads)
    LP(8, 12, false, 0, false, Are, Aim, Bre, Bim, 0, 1);
    // rep 1
    LP(0, 4,  true,  2, false, Bre, Bim, Are, Aim, 1, 0);  // RY (+perm r=2 on loads)
    LP(8, 12, true,  0, false, Are, Aim, Bre, Bim, 1, 0);
    LP(0, 4,  false, 0, false, Bre, Bim, Are, Aim, 1, 0);  // Rot L0
    LP(8, 12, false, 0, false, Are, Aim, Bre, Bim, 1, 0);
    LP(0, 4,  false, 1, false, Bre, Bim, Are, Aim, 1, 1);  // Rot L1 (+perm r=1)
    LP(8, 12, false, 0, false, Are, Aim, Bre, Bim, 1, 1);
    // measurement (+final perm r=2 via inverse map)
    qsim_meas<2><<<g, blk, 0, stream>>>(Bre, Bim, out);
#undef LP
}